// LSTM_45406394253989
// MI455X (gfx1250) — compile-verified
//
#include <hip/hip_runtime.h>
#include <cstdint>
#include <cstddef>

typedef __attribute__((ext_vector_type(2))) float v2f;
typedef __attribute__((ext_vector_type(8))) float v8f;

#define HIDN   1024
#define LSTMH  512
#define GATEN  2048
#define TSEQ   512
#define MROW   64
#define NBATCH 32768
#define KPAD   260     // 257 padded to multiple of 4 (even stride)

__device__ __forceinline__ v8f wmma4(v2f a, v2f b, v8f c) {
  // V_WMMA_F32_16X16X4_F32 : D = A(16x4,f32) x B(4x16,f32) + C(16x16,f32)
  return __builtin_amdgcn_wmma_f32_16x16x4_f32(false, a, false, b, (short)0, c,
                                               false, false);
}

__device__ __forceinline__ float sigmf(float x) {
  return 1.0f / (1.0f + __expf(-x));
}

__device__ __forceinline__ v2f tov2(float2 v) {
  v2f r; r.x = v.x; r.y = v.y; return r;
}

// ---------------------------------------------------------------------------
// Copy rows with zero-padding of the K dimension (for odd-stride operands).
// ---------------------------------------------------------------------------
__global__ __launch_bounds__(256)
void pack_pad_kernel(const float* __restrict__ src, long sstride,
                     int rows, int cols, float* __restrict__ dst, int dstride)
{
  const long i = (long)blockIdx.x * blockDim.x + threadIdx.x;
  if (i >= (long)rows * dstride) return;
  const int r = (int)(i / dstride);
  const int c = (int)(i - (long)r * dstride);
  dst[i] = (c < cols) ? src[(long)r * sstride + c] : 0.0f;
}

// ---------------------------------------------------------------------------
// Generic  C[M,N] = act( A[M,K] (row-major, stride lda) @ W[N,K]^T + b (+b2) )
// One wave computes a 16x64 strip (4 x 16x16 WMMA accumulators).
// Requirements (guaranteed by caller): K % 4 == 0, lda even, K even,
// A and W 8-byte aligned -> branch-free float2 (b64) loads.
// K-loop is software-pipelined (prefetch k+1 while WMMAs consume k) so the
// compiler can issue partial s_wait_loadcnt instead of full drains.
// ---------------------------------------------------------------------------
__global__ __launch_bounds__(256)
void gemm_wtn_kernel(const float* __restrict__ A, long lda,
                     const float* __restrict__ W,
                     const float* __restrict__ bias,
                     const float* __restrict__ bias2,
                     float* __restrict__ C,
                     int M, int N, int K, int relu)
{
  const int lane = threadIdx.x & 31;
  const int l16  = lane & 15;
  const int lhi  = lane >> 4;             // 0 or 1
  const int tiles_n = (N + 63) >> 6;
  const int wid = blockIdx.x * (blockDim.x >> 5) + (threadIdx.x >> 5);
  if (wid >= (M >> 4) * tiles_n) return;
  const int tm = wid / tiles_n;
  const int tn = wid - tm * tiles_n;
  const int m0 = tm << 4;
  const int n0 = tn << 6;

  int neff[4];
  v8f acc[4];
#pragma unroll
  for (int j = 0; j < 4; ++j) {
    const int n = n0 + j * 16 + l16;
    neff[j] = (n < N) ? n : (N - 1);      // clamp: always a valid address
    float bv = bias[neff[j]];
    if (bias2) bv += bias2[neff[j]];
#pragma unroll
    for (int r = 0; r < 8; ++r) acc[j][r] = bv;
  }

  const float* ap  = A + (long)(m0 + l16) * lda + 2 * lhi;
  const float* wp0 = W + (long)neff[0] * K + 2 * lhi;
  const float* wp1 = W + (long)neff[1] * K + 2 * lhi;
  const float* wp2 = W + (long)neff[2] * K + 2 * lhi;
  const float* wp3 = W + (long)neff[3] * K + 2 * lhi;

  // prologue: fragments for k0 = 0
  float2 av  = *(const float2*)ap;
  float2 b0v = *(const float2*)wp0;
  float2 b1v = *(const float2*)wp1;
  float2 b2v = *(const float2*)wp2;
  float2 b3v = *(const float2*)wp3;

#pragma unroll 2
  for (int k0 = 0; k0 < K - 4; k0 += 4) {
    ap += 4; wp0 += 4; wp1 += 4; wp2 += 4; wp3 += 4;
    const float2 avn = *(const float2*)ap;      // prefetch k0+4
    const float2 b0n = *(const float2*)wp0;
    const float2 b1n = *(const float2*)wp1;
    const float2 b2n = *(const float2*)wp2;
    const float2 b3n = *(const float2*)wp3;
    const v2f a = tov2(av);
    acc[0] = wmma4(a, tov2(b0v), acc[0]);       // consume k0
    acc[1] = wmma4(a, tov2(b1v), acc[1]);
    acc[2] = wmma4(a, tov2(b2v), acc[2]);
    acc[3] = wmma4(a, tov2(b3v), acc[3]);
    av = avn; b0v = b0n; b1v = b1n; b2v = b2n; b3v = b3n;
  }
  {
    const v2f a = tov2(av);                     // epilogue: last k-step
    acc[0] = wmma4(a, tov2(b0v), acc[0]);
    acc[1] = wmma4(a, tov2(b1v), acc[1]);
    acc[2] = wmma4(a, tov2(b2v), acc[2]);
    acc[3] = wmma4(a, tov2(b3v), acc[3]);
  }

#pragma unroll
  for (int j = 0; j < 4; ++j) {
    const int n = n0 + j * 16 + l16;
    if (n < N) {
#pragma unroll
      for (int r = 0; r < 8; ++r) {
        float v = acc[j][r];
        if (relu) v = fmaxf(v, 0.0f);
        C[(long)(m0 + r + 8 * lhi) * N + n] = v;
      }
    }
  }
}

// ---------------------------------------------------------------------------
// Persistent recurrent kernel: 8 workgroups x 512 threads (16 waves).
// Workgroup g owns h-columns [64g, 64g+64) of both LSTM layers.
// Wave (mi,ni) owns the 16x16 tile (rows 16mi, cols 16ni) and keeps its
// c0/c1 cell-state tiles in registers for all 512 steps.
// ---------------------------------------------------------------------------
__device__ __forceinline__ void ldscopy64x512(float* __restrict__ dst,
                                              const float* __restrict__ src)
{
  const float4* s = (const float4*)src;
  float4* d = (float4*)dst;
#pragma unroll
  for (int i = 0; i < (MROW * LSTMH / 4) / 512; ++i)
    d[threadIdx.x + i * 512] = s[threadIdx.x + i * 512];
}

__device__ __forceinline__ void kloop4(v8f acc[4],
                                       const float* __restrict__ sh,
                                       const float* __restrict__ W,
                                       int m0, int ncol, int l16, int lhi)
{
  const float* arow = sh + (m0 + l16) * LSTMH + 2 * lhi;
  const float* w0 = W + (size_t)(0 * LSTMH + ncol) * LSTMH + 2 * lhi;
  const float* w1 = W + (size_t)(1 * LSTMH + ncol) * LSTMH + 2 * lhi;
  const float* w2 = W + (size_t)(2 * LSTMH + ncol) * LSTMH + 2 * lhi;
  const float* w3 = W + (size_t)(3 * LSTMH + ncol) * LSTMH + 2 * lhi;

  float2 av  = *(const float2*)arow;            // ds_load_b64
  float2 b0v = *(const float2*)w0;              // global_load_b64 (L2)
  float2 b1v = *(const float2*)w1;
  float2 b2v = *(const float2*)w2;
  float2 b3v = *(const float2*)w3;

#pragma unroll 2
  for (int k0 = 0; k0 < LSTMH - 4; k0 += 4) {
    arow += 4; w0 += 4; w1 += 4; w2 += 4; w3 += 4;
    const float2 avn = *(const float2*)arow;    // prefetch k0+4
    const float2 b0n = *(const float2*)w0;
    const float2 b1n = *(const float2*)w1;
    const float2 b2n = *(const float2*)w2;
    const float2 b3n = *(const float2*)w3;
    const v2f a = tov2(av);
    acc[0] = wmma4(a, tov2(b0v), acc[0]);
    acc[1] = wmma4(a, tov2(b1v), acc[1]);
    acc[2] = wmma4(a, tov2(b2v), acc[2]);
    acc[3] = wmma4(a, tov2(b3v), acc[3]);
    av = avn; b0v = b0n; b1v = b1n; b2v = b2n; b3v = b3n;
  }
  {
    const v2f a = tov2(av);
    acc[0] = wmma4(a, tov2(b0v), acc[0]);
    acc[1] = wmma4(a, tov2(b1v), acc[1]);
    acc[2] = wmma4(a, tov2(b2v), acc[2]);
    acc[3] = wmma4(a, tov2(b3v), acc[3]);
  }
}

__device__ __forceinline__ v8f lstm_cell8(v8f acc[4], v8f& c)
{
  v8f h;
#pragma unroll
  for (int r = 0; r < 8; ++r) {
    const float i_ = sigmf(acc[0][r]);
    const float f_ = sigmf(acc[1][r]);
    const float g_ = tanhf(acc[2][r]);
    const float o_ = sigmf(acc[3][r]);
    const float cn = f_ * c[r] + i_ * g_;
    c[r] = cn;
    h[r] = o_ * tanhf(cn);
  }
  return h;
}

__device__ __forceinline__ void gsync(int* ctr, int target)
{
  __syncthreads();
  if (threadIdx.x == 0) {
    __threadfence();                       // release my stores
    atomicAdd(ctr, 1);
    while (__atomic_load_n(ctr, __ATOMIC_RELAXED) < target)
      __builtin_amdgcn_s_sleep(1);
    __threadfence();                       // acquire others' stores
  }
  __syncthreads();
}

__global__ __launch_bounds__(512)
void lstm_rec_kernel(const float* __restrict__ A1,     // (32768, 2048) bias-folded
                     const float* __restrict__ w1hh,   // (2048, 512)
                     const float* __restrict__ w2ih,   // (2048, 512)
                     const float* __restrict__ w2hh,   // (2048, 512)
                     const float* __restrict__ b2a,
                     const float* __restrict__ b2b,
                     float* __restrict__ h0b,          // 2 * 64*512 (ping-pong)
                     float* __restrict__ h1hist,       // (64+32768) * 512
                     int* __restrict__ ctr)
{
  extern __shared__ float sh[];                        // 64*512 floats (128 KB)
  const int lane = threadIdx.x & 31;
  const int l16  = lane & 15;
  const int lhi  = lane >> 4;
  const int wave = threadIdx.x >> 5;                   // 0..15
  const int mi = wave >> 2, ni = wave & 3;
  const int m0 = mi * 16;
  const int ncol = blockIdx.x * 64 + ni * 16 + l16;    // owned h-column
  const int NWG = (int)gridDim.x;

  float b2g[4];
#pragma unroll
  for (int gi = 0; gi < 4; ++gi)
    b2g[gi] = b2a[gi * LSTMH + ncol] + b2b[gi * LSTMH + ncol];

  v8f c0t = {}; v8f c1t = {};
  int bar = 0;

  for (int t = 0; t < TSEQ; ++t) {
    const float* h0r = h0b + (size_t)(t & 1) * (MROW * LSTMH);
    float*       h0w = h0b + (size_t)((t + 1) & 1) * (MROW * LSTMH);

    // ---------- stage A: LSTM layer 1 ----------
    ldscopy64x512(sh, h0r);
    __syncthreads();

    v8f acc[4];
    const float* A1t = A1 + (size_t)t * (MROW * GATEN);
#pragma unroll
    for (int gi = 0; gi < 4; ++gi) {
#pragma unroll
      for (int r = 0; r < 8; ++r)
        acc[gi][r] =
            A1t[(size_t)(m0 + r + 8 * lhi) * GATEN + gi * LSTMH + ncol];
    }

    kloop4(acc, sh, w1hh, m0, ncol, l16, lhi);

    {
      const v8f hnew = lstm_cell8(acc, c0t);
#pragma unroll
      for (int r = 0; r < 8; ++r)
        h0w[(size_t)(m0 + r + 8 * lhi) * LSTMH + ncol] = hnew[r];
    }

    gsync(ctr, (++bar) * NWG);   // h0(t) visible everywhere

    // ---------- stage B: LSTM layer 2 ----------
#pragma unroll
    for (int gi = 0; gi < 4; ++gi) {
#pragma unroll
      for (int r = 0; r < 8; ++r) acc[gi][r] = b2g[gi];
    }

    ldscopy64x512(sh, h0w);                            // h0 current
    __syncthreads();
    kloop4(acc, sh, w2ih, m0, ncol, l16, lhi);
    __syncthreads();
    ldscopy64x512(sh, h1hist + (size_t)t * (MROW * LSTMH));  // h1 previous
    __syncthreads();
    kloop4(acc, sh, w2hh, m0, ncol, l16, lhi);

    {
      const v8f h1new = lstm_cell8(acc, c1t);
      float* h1o = h1hist + (size_t)(t + 1) * (MROW * LSTMH);
#pragma unroll
      for (int r = 0; r < 8; ++r)
        h1o[(size_t)(m0 + r + 8 * lhi) * LSTMH + ncol] = h1new[r];
    }

    gsync(ctr, (++bar) * NWG);   // h1(t) visible, LDS safe to reuse
  }
}

// ---------------------------------------------------------------------------
extern "C" void kernel_launch(void* const* d_in, const int* in_sizes, int n_in,
                              void* d_out, int out_size, void* d_ws,
                              size_t ws_size, hipStream_t stream)
{
  (void)in_sizes; (void)n_in; (void)out_size; (void)ws_size;
  const float* x      = (const float*)d_in[0];
  const float* fc1_w  = (const float*)d_in[1];
  const float* fc1_b  = (const float*)d_in[2];
  const float* fc2_w  = (const float*)d_in[3];
  const float* fc2_b  = (const float*)d_in[4];
  const float* l1_wih = (const float*)d_in[5];
  const float* l1_whh = (const float*)d_in[6];
  const float* l1_bih = (const float*)d_in[7];
  const float* l1_bhh = (const float*)d_in[8];
  const float* l2_wih = (const float*)d_in[9];
  const float* l2_whh = (const float*)d_in[10];
  const float* l2_bih = (const float*)d_in[11];
  const float* l2_bhh = (const float*)d_in[12];
  const float* fc3_w  = (const float*)d_in[13];
  const float* fc3_b  = (const float*)d_in[14];
  float* out = (float*)d_out;

  float* ws = (float*)d_ws;
  float* z1      = ws;                        // 32768*1024 floats
  float* h1hist  = ws;                        // aliases z1 (z1 dead after fc2)
  float* z2      = ws + 33554432ULL;          // 32768*1024
  float* A1      = ws + 67108864ULL;          // 32768*2048
  float* framep  = A1;                        // 32768*260, dead before A1 write
  float* fc1wp   = A1 + 8519680ULL;           // 1024*260,  dead before A1 write
  float* h0b     = ws + 134217728ULL;         // 2*64*512
  int*   ctr     = (int*)(ws + 134283264ULL);

  auto launch_gemm = [&](const float* A, long lda, const float* W,
                         const float* b, const float* b2, float* C,
                         int M, int N, int K, int relu) {
    const int tiles = (M / 16) * ((N + 63) / 64);
    dim3 grid((unsigned)((tiles + 7) / 8)), block(256);
    gemm_wtn_kernel<<<grid, block, 0, stream>>>(A, lda, W, b, b2, C,
                                                M, N, K, relu);
  };
  auto launch_pack = [&](const float* src, long sstride, int rows, int cols,
                         float* dst, int dstride) {
    const long total = (long)rows * dstride;
    dim3 grid((unsigned)((total + 255) / 256)), block(256);
    pack_pad_kernel<<<grid, block, 0, stream>>>(src, sstride, rows, cols,
                                                dst, dstride);
  };

  // Phase 0: zero-pad odd-stride fc1 operands to K=260 (even, mult of 4).
  // frame = x[:, 4, :]  (mid = 7//2 + 1 = 4)
  launch_pack(x + 4 * 257, 7 * 257, NBATCH, 257, framep, KPAD);
  launch_pack(fc1_w, 257, HIDN, 257, fc1wp, KPAD);

  // Phase 1: parallel over all 32768 rows.
  launch_gemm(framep, KPAD, fc1wp, fc1_b, nullptr, z1, NBATCH, HIDN, KPAD, 0);
  launch_gemm(z1, HIDN, fc2_w, fc2_b, nullptr, z2, NBATCH, HIDN, HIDN, 1);

  // State init (stream-ordered AFTER fc2 consumed z1, since h1hist aliases z1)
  hipMemsetAsync(h1hist, 0, (size_t)MROW * LSTMH * sizeof(float), stream);
  hipMemsetAsync(h0b, 0, (size_t)2 * MROW * LSTMH * sizeof(float), stream);
  hipMemsetAsync(ctr, 0, sizeof(int), stream);

  // A1 = z @ l1_wih^T + (l1_bih + l1_bhh), for all 512 steps at once.
  launch_gemm(z2, HIDN, l1_wih, l1_bih, l1_bhh, A1, NBATCH, GATEN, HIDN, 0);

  // Phase 2: persistent recurrent kernel (512 sequential steps).
  hipFuncSetAttribute(reinterpret_cast<const void*>(&lstm_rec_kernel),
                      hipFuncAttributeMaxDynamicSharedMemorySize,
                      MROW * LSTMH * (int)sizeof(float));
  lstm_rec_kernel<<<dim3(8), dim3(512), MROW * LSTMH * sizeof(float), stream>>>(
      A1, l1_whh, l2_wih, l2_whh, l2_bih, l2_bhh, h0b, h1hist, ctr);

  // Phase 3: out = h1 @ fc3_w^T + fc3_b   (N = 257, edge-tiles bounds-checked)
  launch_gemm(h1hist + (size_t)MROW * LSTMH, LSTMH, fc3_w, fc3_b, nullptr, out,
              NBATCH, 257, LSTMH, 0);
}